// MultiHeadAttention_71562745085969
// MI455X (gfx1250) — compile-verified
//
#include <hip/hip_runtime.h>
#include <math.h>

typedef __attribute__((ext_vector_type(2))) float v2f;
typedef __attribute__((ext_vector_type(4))) float v4f;
typedef __attribute__((ext_vector_type(8))) float v8f;

#define NB      8
#define DMODEL  4096
#define NH      32
#define DHEAD   128
#define LCACHE  4095
#define KTOT    4096   // LCACHE + 1

// ---------------------------------------------------------------------------
// Out(8 x 4096) = A(8 x 4096) @ W^T   (W row-major 4096 x 4096)
// grid = 256 blocks (one 16-column tile), block = 256 threads = 8 waves.
// Wave w accumulates K-slice [512w, 512w+512) with V_WMMA_F32_16X16X4_F32;
// split-K partials reduced through LDS. Only M rows 0..7 (batch) are valid;
// the A upper half is zeroed by multiply so EXEC stays all-ones for WMMA.
// ---------------------------------------------------------------------------
__global__ __launch_bounds__(256)
void gemm_xWt_wmma(const float* __restrict__ A, const float* __restrict__ W,
                   float* __restrict__ Out) {
  const int tid  = threadIdx.x;
  const int wave = tid >> 5;
  const int lane = tid & 31;
  const int m    = lane & 15;   // tile row this lane feeds (A: M index, B: N index)
  const int hh   = lane >> 4;   // 0 -> K pair {0,1}, 1 -> K pair {2,3}
  const int col0 = blockIdx.x * 16;

  const float* arow  = A + (size_t)(m & 7) * DMODEL;          // clamp row; mask below
  const float  amask = (m < NB) ? 1.0f : 0.0f;
  const float* wrow  = W + (size_t)(col0 + m) * DMODEL;       // B[k][n] = W[col0+n][k]

  v8f c = {};
  const int kbeg = wave * 512;
  #pragma unroll 8
  for (int k = kbeg; k < kbeg + 512; k += 4) {
    v2f a = *(const v2f*)(arow + k + 2 * hh);   // A[m][k+2hh .. +1]
    a *= amask;
    v2f b = *(const v2f*)(wrow + k + 2 * hh);   // B frag: W row = col0+m
    c = __builtin_amdgcn_wmma_f32_16x16x4_f32(false, a, false, b,
                                              (short)0, c, false, false);
  }

  // C layout: VGPR i, lanes 0..15 hold C[M=i][N=lane]; rows 0..7 are the batch.
  __shared__ float red[8][NB][16];
  if (lane < 16) {
    #pragma unroll
    for (int i = 0; i < NB; ++i) red[wave][i][lane] = c[i];
  }
  __syncthreads();
  if (tid < NB * 16) {
    const int b = tid >> 4, n = tid & 15;
    float s = 0.f;
    #pragma unroll
    for (int w2 = 0; w2 < 8; ++w2) s += red[w2][b][n];
    Out[(size_t)b * DMODEL + col0 + n] = s;
  }
}

// ---------------------------------------------------------------------------
// Flash-decode attention: one block per (b,h), 8 waves split the 4096 keys.
// Dim-parallel (float4 per lane -> fully coalesced 512B K/V rows), with keys
// processed 8 at a time so the 5-deep shuffle reductions overlap and the
// online-softmax state is updated once per 8-key chunk (1 corr-exp + 8 exps
// per chunk instead of 2 exps + full chain per key). Key LCACHE redirects to
// the freshly projected k/v. K/V prefetched one chunk ahead.
// ---------------------------------------------------------------------------
__global__ __launch_bounds__(256)
void attn_decode(const float* __restrict__ qbuf, const float* __restrict__ kcache,
                 const float* __restrict__ vcache, const float* __restrict__ knew,
                 const float* __restrict__ vnew, float* __restrict__ obuf) {
  const int bh = blockIdx.x;
  const int b  = bh / NH, h = bh % NH;
  const int tid = threadIdx.x, wave = tid >> 5, lane = tid & 31;

  const float scale = 0.0883883476483184405f;   // 1/sqrt(128)
  const float* qp = qbuf + (size_t)b * DMODEL + h * DHEAD;
  v4f q4 = *(const v4f*)(qp + 4 * lane);
  q4 *= scale;

  const size_t headbase = ((size_t)b * NH + h) * (size_t)LCACHE * DHEAD;
  const float* knewp = knew + (size_t)b * DMODEL + h * DHEAD;
  const float* vnewp = vnew + (size_t)b * DMODEL + h * DHEAD;

  float mcur = -__builtin_inff(), lsum = 0.f;
  v4f acc = {};

  const int jbeg = wave * (KTOT / 8);
  const int jend = jbeg + (KTOT / 8);
  for (int j0 = jbeg; j0 < jend; j0 += 8) {
    // Prefetch next chunk's K/V rows (global_prefetch_b8; streaming hint).
    if (j0 + 8 < jend) {
      const size_t pofs = headbase + (size_t)(j0 + 8) * DHEAD + 4 * lane;
      __builtin_prefetch(kcache + pofs, 0, 0);
      __builtin_prefetch(vcache + pofs, 0, 0);
    }

    float p[8];
    v4f   vv[8];
    #pragma unroll
    for (int u = 0; u < 8; ++u) {
      const int j = j0 + u;
      const float* kr = (j < LCACHE) ? (kcache + headbase + (size_t)j * DHEAD) : knewp;
      const float* vr = (j < LCACHE) ? (vcache + headbase + (size_t)j * DHEAD) : vnewp;
      v4f kv = *(const v4f*)(kr + 4 * lane);
      vv[u]  = *(const v4f*)(vr + 4 * lane);
      p[u] = kv.x * q4.x + kv.y * q4.y + kv.z * q4.z + kv.w * q4.w;
    }
    // Eight independent butterfly reductions (overlapping latency).
    #pragma unroll
    for (int u = 0; u < 8; ++u) {
      #pragma unroll
      for (int off = 16; off > 0; off >>= 1) p[u] += __shfl_xor(p[u], off, 32);
    }
    // Chunk max (tree), single state update.
    float cm0 = fmaxf(p[0], p[1]), cm1 = fmaxf(p[2], p[3]);
    float cm2 = fmaxf(p[4], p[5]), cm3 = fmaxf(p[6], p[7]);
    const float cmax = fmaxf(fmaxf(cm0, cm1), fmaxf(cm2, cm3));
    const float mn   = fmaxf(mcur, cmax);
    const float corr = __expf(mcur - mn);

    float e[8];
    #pragma unroll
    for (int u = 0; u < 8; ++u) e[u] = __expf(p[u] - mn);
    const float es = ((e[0] + e[1]) + (e[2] + e[3])) + ((e[4] + e[5]) + (e[6] + e[7]));
    lsum = lsum * corr + es;

    v4f t = vv[0] * e[0];
    #pragma unroll
    for (int u = 1; u < 8; ++u) t += vv[u] * e[u];
    acc  = acc * corr + t;
    mcur = mn;
  }

  __shared__ float sm[8], sl[8];
  __shared__ v4f   saccv[8][32];
  if (lane == 0) { sm[wave] = mcur; sl[wave] = lsum; }
  saccv[wave][lane] = acc;
  __syncthreads();

  if (tid < DHEAD) {
    float M = sm[0];
    #pragma unroll
    for (int w = 1; w < 8; ++w) M = fmaxf(M, sm[w]);
    const float* sa = (const float*)saccv;
    float Ls = 0.f, o = 0.f;
    #pragma unroll
    for (int w = 0; w < 8; ++w) {
      const float e = __expf(sm[w] - M);
      Ls += sl[w] * e;
      o  += sa[w * DHEAD + tid] * e;
    }
    obuf[(size_t)b * DMODEL + h * DHEAD + tid] = o / Ls;
  }
}

// ---------------------------------------------------------------------------
extern "C" void kernel_launch(void* const* d_in, const int* in_sizes, int n_in,
                              void* d_out, int out_size, void* d_ws, size_t ws_size,
                              hipStream_t stream) {
  const float* x  = (const float*)d_in[0];
  const float* kc = (const float*)d_in[1];
  const float* vc = (const float*)d_in[2];
  const float* Wq = (const float*)d_in[3];
  const float* Wk = (const float*)d_in[4];
  const float* Wv = (const float*)d_in[5];
  const float* Wo = (const float*)d_in[6];
  float* out = (float*)d_out;

  float* qb = (float*)d_ws;            // 8*4096 each
  float* kb = qb + NB * DMODEL;
  float* vb = kb + NB * DMODEL;
  float* ab = vb + NB * DMODEL;

  const dim3 blk(256);
  gemm_xWt_wmma<<<dim3(DMODEL / 16), blk, 0, stream>>>(x, Wq, qb);
  gemm_xWt_wmma<<<dim3(DMODEL / 16), blk, 0, stream>>>(x, Wk, kb);
  gemm_xWt_wmma<<<dim3(DMODEL / 16), blk, 0, stream>>>(x, Wv, vb);
  attn_decode<<<dim3(NB * NH), blk, 0, stream>>>(qb, kc, vc, kb, vb, ab);
  gemm_xWt_wmma<<<dim3(DMODEL / 16), blk, 0, stream>>>(ab, Wo, out);
}